// GraphAttentionalLayer_730144440901
// MI455X (gfx1250) — compile-verified
//
#include <hip/hip_runtime.h>
#include <hip/hip_bf16.h>
#include <math.h>

// GAT layer: softmax(si_i + sj_j over j) collapses: si cancels, attention row
// is identical for every i.  out[b,i,:] = r[b,:] = (softmax(sj) @ hW)[b,:].
//
// B=2, N=4096, D=128.  hW = h @ W^T done with V_WMMA_F32_16X16X4_F32 (f32 WMMA).

#define B_ 2
#define N_ 4096
#define D_ 128
#define ROWS_ (B_ * N_)   // 8192

typedef __attribute__((ext_vector_type(2))) float v2f;
typedef __attribute__((ext_vector_type(8))) float v8f;

// ---------------------------------------------------------------------------
// Kernel 1: hW[row][e] = sum_d h[row][d] * W[e][d]   (h @ W^T)
// One wave per 16x16 output tile, K-loop of 32 x wmma_f32_16x16x4_f32.
// A (16x4 f32): lane 0-15 row M=lane, VGPR0/1 = K0/K1; lanes 16-31 K2/K3.
// B (4x16 f32): same layout on B^T -> lane holds W[e=col][k..k+1] contiguous.
// C/D (16x16 f32, 8 VGPRs): vgpr v -> M = v + 8*(lane>=16), N = lane&15.
// ---------------------------------------------------------------------------
__global__ __launch_bounds__(256) void gat_gemm_hw(
    const float* __restrict__ h, const float* __restrict__ W,
    float* __restrict__ hW) {
  const int lane = threadIdx.x & 31;
  const int wave = threadIdx.x >> 5;
  const int tile = blockIdx.x * 8 + wave;   // 4096 tiles total
  const int tm = tile >> 3;                 // 0..511 (row tiles over B*N)
  const int tn = tile & 7;                  // 0..7   (col tiles over D)

  const int arow = tm * 16 + (lane & 15);
  const int bcol = tn * 16 + (lane & 15);
  const int khalf = (lane >> 4) * 2;        // 0 or 2

  const float* __restrict__ hrow = h + (size_t)arow * D_;
  const float* __restrict__ wrow = W + (size_t)bcol * D_;  // W[e][*], e = bcol

  v8f acc = {};
#pragma unroll
  for (int k = 0; k < D_; k += 4) {
    v2f a = *(const v2f*)(hrow + k + khalf);
    v2f b = *(const v2f*)(wrow + k + khalf);
    acc = __builtin_amdgcn_wmma_f32_16x16x4_f32(
        /*neg_a=*/false, a, /*neg_b=*/false, b,
        /*c_mod=*/(short)0, acc, /*reuse_a=*/false, /*reuse_b=*/false);
  }

  const int mbase = tm * 16 + (lane >> 4) * 8;
  const int nout  = tn * 16 + (lane & 15);
#pragma unroll
  for (int v = 0; v < 8; ++v)
    hW[(size_t)(mbase + v) * D_ + nout] = acc[v];
}

// ---------------------------------------------------------------------------
// Kernel 2: sj[row] = dot(hW[row,:], aj)   — one wave per row
// ---------------------------------------------------------------------------
__global__ __launch_bounds__(256) void gat_row_dot(
    const float* __restrict__ hW, const float* __restrict__ aj,
    float* __restrict__ sj) {
  const int lane = threadIdx.x & 31;
  const int wave = threadIdx.x >> 5;
  const int row = blockIdx.x * 8 + wave;
  const float4 x = ((const float4*)(hW + (size_t)row * D_))[lane];
  const float4 a = ((const float4*)aj)[lane];
  float s = x.x * a.x + x.y * a.y + x.z * a.z + x.w * a.w;
#pragma unroll
  for (int off = 16; off; off >>= 1) s += __shfl_xor(s, off, 32);
  if (lane == 0) sj[row] = s;
}

// ---------------------------------------------------------------------------
// Kernel 3: per-batch softmax stats: p_j = exp(sj_j - max), invs = 1/sum(p)
// One block (256 threads) per batch.
// ---------------------------------------------------------------------------
__global__ __launch_bounds__(256) void gat_softmax_stats(
    const float* __restrict__ sj, float* __restrict__ p,
    float* __restrict__ invs) {
  __shared__ float red[256];
  const int b = blockIdx.x;
  const int t = threadIdx.x;
  const float* __restrict__ s = sj + (size_t)b * N_;

  float m = -INFINITY;
  for (int j = t; j < N_; j += 256) m = fmaxf(m, s[j]);
  red[t] = m;
  __syncthreads();
  for (int off = 128; off; off >>= 1) {
    if (t < off) red[t] = fmaxf(red[t], red[t + off]);
    __syncthreads();
  }
  m = red[0];
  __syncthreads();

  float acc = 0.f;
  for (int j = t; j < N_; j += 256) {
    float e = __expf(s[j] - m);
    p[(size_t)b * N_ + j] = e;
    acc += e;
  }
  red[t] = acc;
  __syncthreads();
  for (int off = 128; off; off >>= 1) {
    if (t < off) red[t] += red[t + off];
    __syncthreads();
  }
  if (t == 0) invs[b] = 1.0f / red[0];
}

// ---------------------------------------------------------------------------
// Kernel 4: partial weighted sums over j-chunks of 128.
// grid = B*32 blocks of 128 threads (thread = d), coalesced hW reads.
// ---------------------------------------------------------------------------
#define JCHUNK_ 128
#define NCHUNK_ (N_ / JCHUNK_)   // 32
__global__ __launch_bounds__(128) void gat_wsum_partial(
    const float* __restrict__ hW, const float* __restrict__ p,
    float* __restrict__ part) {
  const int b = blockIdx.x >> 5;
  const int c = blockIdx.x & 31;
  const int d = threadIdx.x;
  const float* __restrict__ base = hW + ((size_t)b * N_ + (size_t)c * JCHUNK_) * D_;
  const float* __restrict__ pp = p + (size_t)b * N_ + (size_t)c * JCHUNK_;
  float acc = 0.f;
#pragma unroll 8
  for (int j = 0; j < JCHUNK_; ++j) acc = fmaf(pp[j], base[(size_t)j * D_ + d], acc);
  part[((size_t)b * NCHUNK_ + c) * D_ + d] = acc;
}

// ---------------------------------------------------------------------------
// Kernel 5: reduce chunks -> r[b,d] = invs[b] * sum_c part[b,c,d]
// ---------------------------------------------------------------------------
__global__ __launch_bounds__(128) void gat_wsum_final(
    const float* __restrict__ part, const float* __restrict__ invs,
    float* __restrict__ r) {
  const int b = blockIdx.x;
  const int d = threadIdx.x;
  float acc = 0.f;
#pragma unroll
  for (int c = 0; c < NCHUNK_; ++c) acc += part[((size_t)b * NCHUNK_ + c) * D_ + d];
  r[(size_t)b * D_ + d] = acc * invs[b];
}

// ---------------------------------------------------------------------------
// Kernel 6: broadcast r[b,:] to every row i of the output (float4 stores).
// out flat float4 index f: row = f>>5, b = row>>12, d4 = f&31.
// ---------------------------------------------------------------------------
__global__ __launch_bounds__(256) void gat_broadcast(
    const float* __restrict__ r, float* __restrict__ out) {
  const int f = blockIdx.x * 256 + threadIdx.x;       // 262144 float4s
  const int b = f >> 17;                               // row>>12 with row=f>>5
  const int d4 = f & 31;
  ((float4*)out)[f] = ((const float4*)r)[b * 32 + d4];
}

// ---------------------------------------------------------------------------
extern "C" void kernel_launch(void* const* d_in, const int* in_sizes, int n_in,
                              void* d_out, int out_size, void* d_ws, size_t ws_size,
                              hipStream_t stream) {
  (void)in_sizes; (void)n_in; (void)out_size; (void)ws_size;
  const float* h  = (const float*)d_in[0];   // [B,N,D]
  const float* W  = (const float*)d_in[1];   // [D,D]
  // d_in[2] = ai : cancels exactly in the softmax -> unused
  const float* aj = (const float*)d_in[3];   // [D]
  float* out = (float*)d_out;

  // Workspace layout (floats)
  float* ws   = (float*)d_ws;
  float* hW   = ws;                              // ROWS_*D_   = 1048576
  float* sj   = hW + (size_t)ROWS_ * D_;         // ROWS_      = 8192
  float* p    = sj + ROWS_;                      // ROWS_      = 8192
  float* part = p + ROWS_;                       // B*32*D_    = 8192
  float* r    = part + (size_t)B_ * NCHUNK_ * D_;// B*D_       = 256
  float* invs = r + (size_t)B_ * D_;             // B_         = 2

  // 1) hW = h @ W^T via f32 WMMA
  gat_gemm_hw<<<(ROWS_ / 16) * (D_ / 16) / 8, 256, 0, stream>>>(h, W, hW);
  // 2) sj = hW . aj
  gat_row_dot<<<ROWS_ / 8, 256, 0, stream>>>(hW, aj, sj);
  // 3) softmax stats per batch
  gat_softmax_stats<<<B_, 256, 0, stream>>>(sj, p, invs);
  // 4) partial weighted sums
  gat_wsum_partial<<<B_ * NCHUNK_, 128, 0, stream>>>(hW, p, part);
  // 5) final reduce + scale
  gat_wsum_final<<<B_, 128, 0, stream>>>(part, invs, r);
  // 6) broadcast to all rows
  gat_broadcast<<<(ROWS_ * D_ / 4) / 256, 256, 0, stream>>>(r, out);
}